// Attention_40802189312014
// MI455X (gfx1250) — compile-verified
//
#include <hip/hip_runtime.h>
#include <math.h>
#include <stdint.h>

typedef __attribute__((ext_vector_type(2))) float v2f;
typedef __attribute__((ext_vector_type(8))) float v8f;

// D = A(16x4) * B(4x16) + C  -- fp32 WMMA, exact precision
#define WMMA_F32(a, b, c) \
  __builtin_amdgcn_wmma_f32_16x16x4_f32(false, (a), false, (b), (short)0, (c), false, false)

#if __has_builtin(__builtin_amdgcn_s_wait_asynccnt)
#define ASYNC_WAIT0() __builtin_amdgcn_s_wait_asynccnt(0)
#else
#define ASYNC_WAIT0() asm volatile("s_wait_asynccnt 0x0" ::: "memory")
#endif

// Per-lane async DMA: 16 bytes global -> LDS, tracked by ASYNCcnt.
__device__ __forceinline__ void async_b128_to_lds(const float* gsrc, const void* lds_dst) {
  unsigned ldsoff = (unsigned)(uintptr_t)lds_dst;  // low 32 bits = LDS offset
  asm volatile("global_load_async_to_lds_b128 %0, %1, off"
               :: "v"(ldsoff), "v"(gsrc) : "memory");
}

constexpr int   BB      = 64;        // batch
constexpr int   DD      = 4096;      // model dim
constexpr int   NH      = 32;        // q heads
constexpr int   KVH     = 8;         // kv heads
constexpr int   HD      = 128;       // head dim
constexpr int   NBLK    = 64;        // blocks per seq
constexpr int   BS      = 16;        // block size
constexpr int   LMAX    = NBLK * BS; // 1024
constexpr int   LPAD    = 132;       // LDS row stride: 528B, 16B aligned, bank-stride 4
constexpr float SM_SCALE = 0.088388347648318447f; // 128^-0.5
constexpr float K_MASK   = -2.3819763e38f;

// ---------------------------------------------------------------------------
// Generic small-M GEMM: C[M x Ncols] = A[M x K] * W
// W element (k, col) at: W[ (col>>7)*cblk + k*KSTR + (col & cmask) ]
//   q_w  (N,D,H):     KSTR=128,  cblk=D*H, cmask=127        (col = n*128+h)
//   kv_w (2,Kv,D,H):  KSTR=128,  cblk=D*H, cmask=127        (col = c*1024+kvh*128+h)
//   o_w  (N,H,D):     KSTR=4096, cblk=0,   cmask=0x7FFFFFFF (col = d, k = n*128+h)
// KSTR is a template constant so ks*KSTR*4 folds into the 24-bit VMEM
// immediate offset: one base pointer per 128-k chunk, zero per-step address math.
// Grid: (Ncols/128, M/16), 256 threads = 8 waves, one 16x16 tile per wave.
// A tile staged into LDS with async b128 DMA; W streamed non-temporally.
// ---------------------------------------------------------------------------
template <int KSTR>
__global__ __launch_bounds__(256)
void gemm_wmma_f32(const float* __restrict__ A, const float* __restrict__ W,
                   float* __restrict__ C, int K, int Ncols,
                   long cblk, int cmask)
{
  __shared__ float lA[16][LPAD];

  const int tid  = threadIdx.x;
  const int wave = tid >> 5;
  const int lane = tid & 31;
  const int hl   = lane & 15;
  const int koff = (lane >> 4) << 1; // 0 or 2

  const int rowbase = blockIdx.y << 4;
  const int col     = (blockIdx.x << 7) + (wave << 4) + hl;
  const float* wp   = W + (long)(col >> 7) * cblk + (col & cmask) + (long)koff * KSTR;

  v8f acc = {};
  for (int k0 = 0; k0 < K; k0 += 128) {
    // async stage A[rowbase..+16][k0..+128]: 512 b128 chunks, 2 per thread
    #pragma unroll
    for (int j = 0; j < 2; ++j) {
      int c  = tid + (j << 8);            // 0..511
      int r  = c >> 5;                    // row 0..15
      int kk = (c & 31) << 2;             // 0,4,..,124
      async_b128_to_lds(A + (size_t)(rowbase + r) * K + (k0 + kk), &lA[r][kk]);
    }
    ASYNC_WAIT0();
    __syncthreads();

    const float* wc = wp + (long)k0 * KSTR;   // single base per chunk
    #pragma unroll
    for (int ks = 0; ks < 128; ks += 4) {
      v2f a, b;
      a.x = lA[hl][ks + koff];
      a.y = lA[hl][ks + koff + 1];
      b.x = __builtin_nontemporal_load(wc + (long)ks * KSTR);        // imm offset
      b.y = __builtin_nontemporal_load(wc + (long)(ks + 1) * KSTR);  // imm offset
      acc = WMMA_F32(a, b, acc);
    }
    __syncthreads();
  }

  const int rsel = (lane >> 4) << 3;
  #pragma unroll
  for (int i = 0; i < 8; ++i) {
    int r = rowbase + i + rsel;
    C[(size_t)r * Ncols + col] = acc[i];
  }
}

// ---------------------------------------------------------------------------
// RoPE on q (in-place) and k, then scatter new k/v rows into output cache.
// One block per batch. pos = context_lens[b]-1.
// kvbuf row layout per b (2048): [key: kvh*128+h | 1024 + value: kvh*128+h]
// ---------------------------------------------------------------------------
__global__ __launch_bounds__(256)
void rope_scatter(float* __restrict__ qbuf, const float* __restrict__ kvbuf,
                  const int* __restrict__ ctxlen, const int* __restrict__ slot_map,
                  float* __restrict__ kcache_out, float* __restrict__ vcache_out)
{
  const int b = blockIdx.x;
  const int t = threadIdx.x;
  const float pos = (float)(ctxlen[b] - 1);
  const int slot = slot_map[b];

  for (int i = t; i < NH * 64; i += 256) {          // q rotate
    int n = i >> 6, j = i & 63;
    float ang = pos * __powf(10000.0f, -(float)j * (1.0f / 64.0f));
    float s = __sinf(ang), c = __cosf(ang);
    float* p = qbuf + ((size_t)b * NH + n) * HD;
    float f = p[j], g = p[j + 64];
    p[j]      = f * c - g * s;
    p[j + 64] = g * c + f * s;
  }
  for (int i = t; i < KVH * 64; i += 256) {         // k rotate + scatter
    int n = i >> 6, j = i & 63;
    float ang = pos * __powf(10000.0f, -(float)j * (1.0f / 64.0f));
    float s = __sinf(ang), c = __cosf(ang);
    const float* p = kvbuf + (size_t)b * 2048 + n * 128;
    float f = p[j], g = p[j + 64];
    float* kd = kcache_out + ((size_t)slot * KVH + n) * HD;
    kd[j]      = f * c - g * s;
    kd[j + 64] = g * c + f * s;
  }
  for (int i = t; i < KVH * HD; i += 256) {         // v scatter
    int n = i >> 7, h = i & 127;
    vcache_out[((size_t)slot * KVH + n) * HD + h] =
        kvbuf[(size_t)b * 2048 + 1024 + n * 128 + h];
  }
}

// ---------------------------------------------------------------------------
// Paged attention. One workgroup per (b, kvh); 8 waves.
//   Phase 1: each wave async-stages a private 16x128 K tile (one page block,
//            coalesced 512B rows) then WMMA Q.K^T from LDS -> logits.
//   Phase 2: waves 0-3 softmax rows via wave32 shuffles.
//   Phase 3: V tiles staged cooperatively (shared by all 8 waves), WMMA P.V.
// ---------------------------------------------------------------------------
__global__ __launch_bounds__(256)
void attn_wmma(const float* __restrict__ qbuf,
               const float* __restrict__ kcache, const float* __restrict__ vcache,
               const int* __restrict__ btab, const int* __restrict__ ctxlen,
               float* __restrict__ outbuf)
{
  __shared__ float qs[16][LPAD];        // 8.4 KB
  __shared__ float probs[4][LMAX];      // 16 KB
  __shared__ float kt[8][16][LPAD];     // 67.6 KB: per-wave K tiles / shared V tile

  const int b    = blockIdx.x >> 3;
  const int kvh  = blockIdx.x & 7;
  const int tid  = threadIdx.x;
  const int wave = tid >> 5;
  const int lane = tid & 31;
  const int hl   = lane & 15;
  const int koff = (lane >> 4) << 1;
  const int rsel = (lane >> 4) << 3;

  const int ctx    = ctxlen[b];
  const int Ltiles = (ctx + 15) >> 4;
  const int Lcap   = Ltiles << 4;
  const int* bt    = btab + b * NBLK;

  // stage q rows (pre-scaled); rows 4..15 zero
  for (int i = tid; i < 16 * HD; i += 256) {
    int r = i >> 7, h = i & 127;
    float v = 0.0f;
    if (r < 4) v = qbuf[((size_t)b * NH + (kvh << 2) + r) * HD + h] * SM_SCALE;
    qs[r][h] = v;
  }
  __syncthreads();

  // ---- Phase 1: Q . K^T ----
  for (int tile = wave; tile < Ltiles; tile += 8) {
    const int slotbase = bt[tile] * BS;
    // wave-private async stage: 16 rows x 512B, one b128 per lane per row
    #pragma unroll
    for (int r = 0; r < 16; ++r) {
      async_b128_to_lds(
          kcache + ((size_t)(slotbase + r) * KVH + kvh) * HD + (lane << 2),
          &kt[wave][r][lane << 2]);
    }
    ASYNC_WAIT0();   // wave-private tile: own ASYNCcnt suffices, no barrier

    v8f acc = {};
    #pragma unroll 8
    for (int ks = 0; ks < HD; ks += 4) {
      v2f a, bf;
      a.x  = qs[hl][ks + koff];
      a.y  = qs[hl][ks + koff + 1];
      bf.x = kt[wave][hl][ks + koff];
      bf.y = kt[wave][hl][ks + koff + 1];
      acc = WMMA_F32(a, bf, acc);
    }
    #pragma unroll
    for (int i = 0; i < 8; ++i) {
      int m = i + rsel;
      if (m < 4) {
        int l = (tile << 4) + hl;
        probs[m][l] = (l < ctx) ? acc[i] : K_MASK;
      }
    }
  }
  __syncthreads();

  // ---- Phase 2: softmax (waves 0-3, one row each) ----
  if (wave < 4) {
    const int g = wave;
    float mx = K_MASK;
    for (int l = lane; l < Lcap; l += 32) mx = fmaxf(mx, probs[g][l]);
    #pragma unroll
    for (int off = 16; off; off >>= 1) mx = fmaxf(mx, __shfl_xor(mx, off, 32));
    float sum = 0.0f;
    for (int l = lane; l < Lcap; l += 32) {
      float e = __expf(probs[g][l] - mx);
      probs[g][l] = e;
      sum += e;
    }
    #pragma unroll
    for (int off = 16; off; off >>= 1) sum += __shfl_xor(sum, off, 32);
    float inv = 1.0f / sum;
    for (int l = lane; l < Lcap; l += 32) probs[g][l] *= inv;
  }
  __syncthreads();

  // ---- Phase 3: P . V ---- (wave w -> h columns [w*16, w*16+16))
  {
    const int hbase = wave << 4;
    v8f acc = {};
    for (int tile = 0; tile < Ltiles; ++tile) {
      const int slotbase = bt[tile] * BS;
      // cooperative stage of V tile (16 x 128): 512 b128 chunks, 2 per thread
      #pragma unroll
      for (int j = 0; j < 2; ++j) {
        int c  = tid + (j << 8);
        int r  = c >> 5;
        int kk = (c & 31) << 2;
        async_b128_to_lds(
            vcache + ((size_t)(slotbase + r) * KVH + kvh) * HD + kk,
            &kt[0][r][kk]);
      }
      ASYNC_WAIT0();
      __syncthreads();

      const int lb = tile << 4;
      #pragma unroll
      for (int ks = 0; ks < 16; ks += 4) {
        v2f a, bf;
        float a0 = 0.0f, a1 = 0.0f;
        if (hl < 4) {
          a0 = probs[hl][lb + ks + koff];
          a1 = probs[hl][lb + ks + koff + 1];
        }
        a.x = a0; a.y = a1;
        bf.x = kt[0][ks + koff][hbase + hl];
        bf.y = kt[0][ks + koff + 1][hbase + hl];
        acc = WMMA_F32(a, bf, acc);
      }
      __syncthreads();
    }
    #pragma unroll
    for (int i = 0; i < 8; ++i) {
      int m = i + rsel;
      if (m < 4)
        outbuf[((size_t)b * NH + (kvh << 2) + m) * HD + hbase + hl] = acc[i];
    }
  }
}

// ---------------------------------------------------------------------------
extern "C" void kernel_launch(void* const* d_in, const int* in_sizes, int n_in,
                              void* d_out, int out_size, void* d_ws, size_t ws_size,
                              hipStream_t stream) {
  (void)in_sizes; (void)n_in; (void)out_size; (void)ws_size;
  const float* x            = (const float*)d_in[0];
  const int*   slot_mapping = (const int*)d_in[2];
  const int*   block_tables = (const int*)d_in[3];
  const int*   context_lens = (const int*)d_in[4];
  const float* cache        = (const float*)d_in[5];
  const float* q_w          = (const float*)d_in[6];
  const float* kv_w         = (const float*)d_in[7];
  const float* o_w          = (const float*)d_in[8];

  float* out = (float*)d_out;
  const size_t side  = (size_t)65536 * KVH * HD;   // one cache side
  float* kcache_out  = out;
  float* vcache_out  = out + side;
  float* out_attn    = out + 2 * side;             // (64, 4096)

  float* qbuf  = (float*)d_ws;                     // 64*32*128
  float* kvbuf = qbuf + BB * NH * HD;              // 64*2048
  float* obuf  = kvbuf + BB * 2 * KVH * HD;        // 64*32*128

  // 1) new_cache = cache (then scatter updates)
  hipMemcpyAsync(out, cache, 2 * side * sizeof(float), hipMemcpyDeviceToDevice, stream);

  // 2) q = x @ q_w   (M=64, K=4096, Ncols=4096)
  gemm_wmma_f32<HD><<<dim3(4096 / 128, BB / 16), 256, 0, stream>>>(
      x, q_w, qbuf, DD, 4096, (long)DD * HD, 127);

  // 3) [k|v] = x @ kv_w (M=64, K=4096, Ncols=2048)
  gemm_wmma_f32<HD><<<dim3(2048 / 128, BB / 16), 256, 0, stream>>>(
      x, kv_w, kvbuf, DD, 2048, (long)DD * HD, 127);

  // 4) RoPE + scatter into cache
  rope_scatter<<<BB, 256, 0, stream>>>(qbuf, kvbuf, context_lens, slot_mapping,
                                       kcache_out, vcache_out);

  // 5) paged attention
  attn_wmma<<<BB * KVH, 256, 0, stream>>>(qbuf, kcache_out, vcache_out,
                                          block_tables, context_lens, obuf);

  // 6) attn_output = obuf @ o_w  (W(k,col) = o_w[k*4096 + col])
  gemm_wmma_f32<DD><<<dim3(4096 / 128, BB / 16), 256, 0, stream>>>(
      obuf, o_w, out_attn, NH * HD, DD, 0L, 0x7FFFFFFF);
}